// HybridGNN_76364518523115
// MI455X (gfx1250) — compile-verified
//
#include <hip/hip_runtime.h>
#include <math.h>

// ---------------------------------------------------------------------------
// HybridGNN for MI455X (gfx1250, wave32).
// All GEMMs ([4000,500]x[500,500] gate matmul, QK^T, GNN diffusion + dense
// layers) run on V_WMMA_F32_16X16X4_F32 (native f32 WMMA): total matrix work
// is only ~5 GFLOP so the f32 matrix pipe is free, and the many sigmoid/elu
// gates + l2 norms want full precision. Row/column normalizations are folded
// into WMMA fragment loads / epilogues so the [B,500,500] attention matrix
// and the Laplacian are never materialized.
// Round 2: branch-free clamped fragment loads (no exec-mask churn), b64
// loads for K-contiguous fragments, unrolled K loops for load batching.
// ---------------------------------------------------------------------------

typedef __attribute__((ext_vector_type(2))) float v2f;
typedef __attribute__((ext_vector_type(8))) float v8f;

#define BN 8
#define TN 64
#define MN 500
#define NH 32
#define DD 160
#define FC 512
#define NODES (BN * MN)  // 4000

__device__ __forceinline__ v8f wmma4(v2f a, v2f b, v8f c) {
  // D = A(16x4,f32) * B(4x16,f32) + C(16x16,f32)
  return __builtin_amdgcn_wmma_f32_16x16x4_f32(false, a, false, b, (short)0, c,
                                               false, false);
}
__device__ __forceinline__ v2f ld2(const float* __restrict__ p) {
  float2 t = *reinterpret_cast<const float2*>(p);  // global_load_b64
  v2f r; r.x = t.x; r.y = t.y; return r;
}
__device__ __forceinline__ float sigm(float x) { return 1.f / (1.f + __expf(-x)); }
__device__ __forceinline__ float eluf(float x) { return x > 0.f ? x : (__expf(x) - 1.f); }

// ---------------- conv (short+long) + GRU, one wave per node ---------------
__global__ void conv_gru_kernel(const float* __restrict__ x,
                                const float* __restrict__ csw, const float* __restrict__ csb,
                                const float* __restrict__ clw, const float* __restrict__ clb,
                                const float* __restrict__ Wih, const float* __restrict__ Whh,
                                const float* __restrict__ bih, const float* __restrict__ bhh,
                                float* __restrict__ hSC, float* __restrict__ lastH) {
  __shared__ float WhhT[3][32][32];  // transposed for conflict-free broadcast
  __shared__ float xs[4][64];
  const int tid = threadIdx.x;
  for (int idx = tid; idx < 3 * 32 * 32; idx += 128) {
    int g = idx >> 10, r = (idx >> 5) & 31, k = idx & 31;
    WhhT[g][k][r] = Whh[(g * 32 + r) * 32 + k];
  }
  const int wave = tid >> 5, lane = tid & 31;
  const int node = blockIdx.x * 4 + wave;  // grid sized exactly: 1000*4 = 4000
  const int b = node / MN, m = node % MN;
  xs[wave][lane]      = x[(b * TN + lane) * MN + m];
  xs[wave][lane + 32] = x[(b * TN + lane + 32) * MN + m];
  __syncthreads();
  const float* xw = xs[wave];

  // conv1d: short = 16 taps-of-64 'VALID' (out len 1); long = dilated-2 (out len 2, mean)
  float hval;
  if (lane < 16) {
    float acc = csb[lane];
    for (int t = 0; t < 64; ++t) acc += xw[t] * csw[lane * 64 + t];
    hval = acc;
  } else {
    int k = lane - 16;
    float a0 = 0.f, a1 = 0.f;
    for (int i = 0; i < 32; ++i) {
      float w = clw[k * 32 + i];
      a0 += xw[2 * i] * w;
      a1 += xw[2 * i + 1] * w;
    }
    hval = 0.5f * (a0 + a1) + clb[k];
  }
  hval = fmaxf(hval, 0.f);  // relu
  hSC[node * 32 + lane] = hval;

  // GRU (input size 1, hidden 32): lane j owns h[j]; broadcast h via shuffles
  const float wr = Wih[lane], wz = Wih[32 + lane], wn = Wih[64 + lane];
  const float bir = bih[lane], biz = bih[32 + lane], bin_ = bih[64 + lane];
  const float bhr = bhh[lane], bhz = bhh[32 + lane], bhn = bhh[64 + lane];
  float h = 0.f;
  for (int t = 0; t < 64; ++t) {
    float xt = xw[t];
    float gr = bhr, gz = bhz, gn = bhn;
    for (int k = 0; k < 32; ++k) {
      float hk = __shfl(h, k, 32);
      gr += WhhT[0][k][lane] * hk;
      gz += WhhT[1][k][lane] * hk;
      gn += WhhT[2][k][lane] * hk;
    }
    float r = sigm(xt * wr + bir + gr);
    float z = sigm(xt * wz + biz + gz);
    float n = tanhf(xt * wn + bin_ + r * gn);
    h = (1.f - z) * n + z * h;
  }
  lastH[node * 32 + lane] = h;
}

// ---------------- Q/K projection (32 -> 64) --------------------------------
__global__ void qk_kernel(const float* __restrict__ hSC,
                          const float* __restrict__ WQw, const float* __restrict__ WQb,
                          const float* __restrict__ WKw, const float* __restrict__ WKb,
                          float* __restrict__ Q, float* __restrict__ Kt) {
  int idx = blockIdx.x * blockDim.x + threadIdx.x;
  if (idx >= NODES * 64) return;
  int node = idx >> 6, o = idx & 63;
  const float* hs = hSC + node * 32;
  float q = WQb[o], k = WKb[o];
  for (int c = 0; c < 32; ++c) {
    float h = hs[c];
    q += h * WQw[o * 32 + c];
    k += h * WKw[o * 32 + c];
  }
  Q[node * 64 + o] = q;
  Kt[node * 64 + o] = k;
}

__global__ void zero_kernel(float* __restrict__ p, int n) {
  int i = blockIdx.x * blockDim.x + threadIdx.x;
  if (i < n) p[i] = 0.f;
}

// ------- S = Q K^T per batch; accumulate per-row sum & sumsq (WMMA) --------
__global__ void attn_stats_kernel(const float* __restrict__ Q, const float* __restrict__ Kt,
                                  float* __restrict__ Ssum, float* __restrict__ Ssq) {
  const int wave = threadIdx.x >> 5, lane = threadIdx.x & 31;
  const int jt = blockIdx.x * 4 + wave;  // 0..31
  const int it = blockIdx.y, b = blockIdx.z;
  const int half = lane >> 4, l16 = lane & 15;
  const int i0 = it * 16, j0 = jt * 16;
  const int mA = i0 + l16, nB = j0 + l16;
  // clamp OOB lanes to a valid address; garbage lands only in C rows/cols
  // that the epilogue discards (invalid cols are zeroed before reduction).
  const int rowA = mA < MN ? mA : MN - 1;
  const int colB = nB < MN ? nB : MN - 1;
  const bool nValid = nB < MN;
  const float* __restrict__ Arow = Q + (size_t)b * MN * 64 + rowA * 64;
  const float* __restrict__ Brow = Kt + (size_t)b * MN * 64 + colB * 64;  // B[k][n]=Kt[n,k]
  v8f acc = {0.f, 0.f, 0.f, 0.f, 0.f, 0.f, 0.f, 0.f};
#pragma unroll
  for (int k0 = 0; k0 < 64; k0 += 4) {
    int ks = k0 + 2 * half;
    acc = wmma4(ld2(Arow + ks), ld2(Brow + ks), acc);  // both K-contiguous -> b64
  }
#pragma unroll
  for (int v = 0; v < 8; ++v) {
    float s = nValid ? acc[v] : 0.f;
    float s2 = s * s;
    for (int off = 8; off; off >>= 1) {  // reduce over the 16 N-lanes of this row
      s += __shfl_xor(s, off, 16);
      s2 += __shfl_xor(s2, off, 16);
    }
    int row = i0 + v + 8 * half;
    if (l16 == 0 && row < MN) {
      atomicAdd(&Ssum[b * MN + row], s);
      atomicAdd(&Ssq[b * MN + row], s2);
    }
  }
}

// ------- assemble feats[0:160] and last_hidden[480:512] into final --------
__global__ void assemble_kernel(const float* __restrict__ hSC,
                                const float* __restrict__ Ssum, const float* __restrict__ Ssq,
                                const float* __restrict__ degree,
                                const float* __restrict__ tencw, const float* __restrict__ tencb,
                                const float* __restrict__ sencw, const float* __restrict__ sencb,
                                const float* __restrict__ lastH, float* __restrict__ fin) {
  int idx = blockIdx.x * blockDim.x + threadIdx.x;
  if (idx >= NODES * 192) return;
  int node = idx / 192, d = idx % 192;
  int m = node % MN;
  float v; int col;
  if (d < 32) { v = hSC[node * 32 + d]; col = d; }
  else if (d < 96) {
    int o = d - 32;
    float rn = Ssum[node] / fmaxf(sqrtf(Ssq[node]), 1e-12f);  // rowsum of l2-normed attn
    v = rn * tencw[o] + tencb[o]; col = d;
  } else if (d < 160) {
    int o = d - 96;
    v = degree[m] * sencw[o] + sencb[o]; col = d;
  } else {
    int o = d - 160; v = lastH[node * 32 + o]; col = 480 + o;
  }
  fin[node * FC + col] = v;
}

// ---------------- A1 = lh @ W1^T, A2 = lh @ W2^T ---------------------------
__global__ void a1a2_kernel(const float* __restrict__ lastH,
                            const float* __restrict__ W1, const float* __restrict__ W2,
                            float* __restrict__ A1, float* __restrict__ A2) {
  int idx = blockIdx.x * blockDim.x + threadIdx.x;
  if (idx >= NODES * NH) return;
  int node = idx >> 5, h = idx & 31;
  const float* lh = lastH + node * 32;
  float s1 = 0.f, s2 = 0.f;
  for (int c = 0; c < 32; ++c) {
    float l = lh[c];
    s1 += l * W1[h * 32 + c];
    s2 += l * W2[h * 32 + c];
  }
  A1[idx] = s1; A2[idx] = s2;
}

// -------- a_raw[b,i,j] = V . elu(A1_i + A2_j + b1) + bv --------------------
__global__ void araw_kernel(const float* __restrict__ A1, const float* __restrict__ A2,
                            const float* __restrict__ b1, const float* __restrict__ V,
                            const float* __restrict__ bv, float* __restrict__ araw) {
  int idx = blockIdx.x * blockDim.x + threadIdx.x;
  if (idx >= BN * MN * MN) return;
  int b = idx / (MN * MN);
  int r = idx - b * MN * MN;
  int i = r / MN, j = r % MN;
  const float* a1 = A1 + (b * MN + i) * 32;
  const float* a2 = A2 + (b * MN + j) * 32;
  float acc = 0.f;
#pragma unroll
  for (int h = 0; h < 32; ++h) acc += eluf(a1[h] + a2[h] + b1[h]) * V[h];
  araw[idx] = acc + bv[0];
}

// -------- column (axis=1) l2 norms of a: rcpn[b,j] -------------------------
__global__ void colnorm_kernel(const float* __restrict__ araw, float* __restrict__ rcpn) {
  __shared__ float red[128];
  int b = blockIdx.x / MN, j = blockIdx.x % MN;
  float s = 0.f;
  for (int i = threadIdx.x; i < MN; i += 128) {
    float v = araw[b * MN * MN + i * MN + j];
    s += v * v;
  }
  red[threadIdx.x] = s;
  __syncthreads();
  for (int off = 64; off; off >>= 1) {
    if (threadIdx.x < off) red[threadIdx.x] += red[threadIdx.x + off];
    __syncthreads();
  }
  if (threadIdx.x == 0) rcpn[blockIdx.x] = 1.f / fmaxf(sqrtf(red[0]), 1e-12f);
}

// ---- WMMA: cg = sigmoid(a_norm @ Wb + wb); full dynamic-adjacency epilogue,
// ---- write binarized adjacency + accumulate row degrees -------------------
__global__ void cgate_adj_kernel(const float* __restrict__ araw, const float* __restrict__ rcpn,
                                 const float* __restrict__ Wb, const float* __restrict__ adj_geo,
                                 const float* __restrict__ d_gate, const float* __restrict__ degree,
                                 const float* __restrict__ wb_p,
                                 float* __restrict__ binb, float* __restrict__ rowdeg) {
  const int wave = threadIdx.x >> 5, lane = threadIdx.x & 31;
  const int jt = blockIdx.x * 4 + wave;  // 0..31
  const int it = blockIdx.y, b = blockIdx.z;
  const int half = lane >> 4, l16 = lane & 15;
  const int i0 = it * 16, j0 = jt * 16;
  const float* __restrict__ A = araw + (size_t)b * MN * MN;
  const float* __restrict__ rn = rcpn + b * MN;
  const int mA = i0 + l16, nB = j0 + l16;
  const int rowA = mA < MN ? mA : MN - 1;
  const int colB = nB < MN ? nB : MN - 1;
  const float* __restrict__ Arow = A + rowA * MN;
  const float* __restrict__ Bcol = Wb + colB;
  v8f acc = {0.f, 0.f, 0.f, 0.f, 0.f, 0.f, 0.f, 0.f};
#pragma unroll 5
  for (int k0 = 0; k0 < MN; k0 += 4) {  // 500 = 125*4
    int ks = k0 + 2 * half;
    v2f a = ld2(Arow + ks);
    v2f r2 = ld2(rn + ks);
    a.x *= r2.x; a.y *= r2.y;            // fold column l2-norm of a
    v2f bb;
    bb.x = Bcol[ks * MN];
    bb.y = Bcol[(ks + 1) * MN];
    acc = wmma4(a, bb, acc);
  }
  const float wb = wb_p[0];
#pragma unroll
  for (int v = 0; v < 8; ++v) {
    int i = i0 + v + 8 * half, j = j0 + l16;
    float binv = 0.f;
    if (i < MN && j < MN) {
      float cg = sigm(acc[v] + wb);
      float an = A[i * MN + j] * rn[j];
      float ag = adj_geo[i * MN + j];
      float dm = sigm(d_gate[i * MN + j] * degree[i] * degree[j]);
      float adjf = ag * cg + an * (1.f - cg) + dm * ag;
      binv = (adjf > 0.f) ? 1.f : adjf;
      binb[(size_t)b * MN * MN + i * MN + j] = binv;
    }
    float s = binv;
    for (int off = 8; off; off >>= 1) s += __shfl_xor(s, off, 16);
    if (l16 == 0 && i < MN) atomicAdd(&rowdeg[b * MN + i], s);
  }
}

__global__ void rcpdeg_kernel(const float* __restrict__ rowdeg, float* __restrict__ rcpdeg) {
  int i = blockIdx.x * blockDim.x + threadIdx.x;
  if (i >= BN * MN) return;
  float d = rowdeg[i];
  rcpdeg[i] = d > 0.f ? 1.f / d : 0.f;  // where(deg>0, 1/deg, 0)
}

// ---- WMMA: tmp = (D^-1) * (bin @ feats); feats read from final buffer -----
__global__ void diffuse_kernel(const float* __restrict__ binb, const float* __restrict__ rcpdeg,
                               const float* __restrict__ fin, int colOff,
                               float* __restrict__ tmp) {
  const int wave = threadIdx.x >> 5, lane = threadIdx.x & 31;
  const int jt = blockIdx.x * 4 + wave;  // d tiles: 0..9
  if (jt >= DD / 16) return;
  const int it = blockIdx.y, b = blockIdx.z;
  const int half = lane >> 4, l16 = lane & 15;
  const int i0 = it * 16, j0 = jt * 16;
  const int mA = i0 + l16, nB = j0 + l16;  // nB < 160 always
  const int rowA = mA < MN ? mA : MN - 1;
  const float* __restrict__ Arow = binb + (size_t)b * MN * MN + rowA * MN;
  const float* __restrict__ Fcol = fin + (size_t)b * MN * FC + colOff + nB;
  v8f acc = {0.f, 0.f, 0.f, 0.f, 0.f, 0.f, 0.f, 0.f};
#pragma unroll 5
  for (int k0 = 0; k0 < MN; k0 += 4) {
    int ks = k0 + 2 * half;
    v2f bb;
    bb.x = Fcol[ks * FC];
    bb.y = Fcol[(ks + 1) * FC];
    acc = wmma4(ld2(Arow + ks), bb, acc);
  }
#pragma unroll
  for (int v = 0; v < 8; ++v) {
    int i = i0 + v + 8 * half, d = j0 + l16;
    if (i < MN) tmp[((size_t)b * MN + i) * DD + d] = acc[v] * rcpdeg[b * MN + i];
  }
}

// ---- WMMA: final[:, dst:dst+160] = elu(tmp @ W + bias) --------------------
__global__ void dense_kernel(const float* __restrict__ tmp, const float* __restrict__ W,
                             const float* __restrict__ bias, float* __restrict__ fin,
                             int dstOff) {
  const int wave = threadIdx.x >> 5, lane = threadIdx.x & 31;
  const int jt = blockIdx.x * 4 + wave;
  if (jt >= DD / 16) return;
  const int it = blockIdx.y, b = blockIdx.z;
  const int half = lane >> 4, l16 = lane & 15;
  const int i0 = it * 16, j0 = jt * 16;
  const int mA = i0 + l16, nB = j0 + l16;  // nB < 160 always
  const int rowA = mA < MN ? mA : MN - 1;
  const float* __restrict__ Arow = tmp + ((size_t)b * MN + rowA) * DD;
  const float* __restrict__ Wcol = W + nB;
  v8f acc = {0.f, 0.f, 0.f, 0.f, 0.f, 0.f, 0.f, 0.f};
#pragma unroll 8
  for (int k0 = 0; k0 < DD; k0 += 4) {  // 160 = 40*4
    int ks = k0 + 2 * half;
    v2f bb;
    bb.x = Wcol[ks * DD];
    bb.y = Wcol[(ks + 1) * DD];
    acc = wmma4(ld2(Arow + ks), bb, acc);
  }
#pragma unroll
  for (int v = 0; v < 8; ++v) {
    int i = i0 + v + 8 * half, o = j0 + l16;
    if (i < MN) fin[((size_t)b * MN + i) * FC + dstOff + o] = eluf(acc[v] + bias[o]);
  }
}

// ---------------- out = final @ out_w^T + out_b ----------------------------
__global__ void out_kernel(const float* __restrict__ fin, const float* __restrict__ ow,
                           const float* __restrict__ ob, float* __restrict__ out) {
  int idx = blockIdx.x * blockDim.x + threadIdx.x;
  if (idx >= NODES) return;
  const float* f = fin + (size_t)idx * FC;
  float acc = ob[0];
  for (int d = 0; d < FC; ++d) acc += f[d] * ow[d];
  out[idx] = acc;
}

// ---------------------------------------------------------------------------
extern "C" void kernel_launch(void* const* d_in, const int* in_sizes, int n_in,
                              void* d_out, int out_size, void* d_ws, size_t ws_size,
                              hipStream_t stream) {
  const float* x        = (const float*)d_in[0];
  const float* adj_geo  = (const float*)d_in[1];
  const float* degree   = (const float*)d_in[2];
  const float* conv_s_w = (const float*)d_in[3];
  const float* conv_s_b = (const float*)d_in[4];
  const float* conv_l_w = (const float*)d_in[5];
  const float* conv_l_b = (const float*)d_in[6];
  const float* WQ_w     = (const float*)d_in[7];
  const float* WQ_b     = (const float*)d_in[8];
  const float* WK_w     = (const float*)d_in[9];
  const float* WK_b     = (const float*)d_in[10];
  const float* tenc_w   = (const float*)d_in[11];
  const float* tenc_b   = (const float*)d_in[12];
  const float* senc_w   = (const float*)d_in[13];
  const float* senc_b   = (const float*)d_in[14];
  const float* gru_Wih  = (const float*)d_in[15];
  const float* gru_Whh  = (const float*)d_in[16];
  const float* gru_bih  = (const float*)d_in[17];
  const float* gru_bhh  = (const float*)d_in[18];
  const float* Vv       = (const float*)d_in[19];
  const float* W1       = (const float*)d_in[20];
  const float* W2       = (const float*)d_in[21];
  const float* Wb       = (const float*)d_in[22];
  const float* bv       = (const float*)d_in[23];
  const float* b1       = (const float*)d_in[24];
  const float* wb       = (const float*)d_in[25];
  const float* d_gate   = (const float*)d_in[26];
  const float* gnn_w    = (const float*)d_in[27];
  const float* gnn_b    = (const float*)d_in[28];
  const float* out_w    = (const float*)d_in[29];
  const float* out_b    = (const float*)d_in[30];

  float* ws = (float*)d_ws;
  float* hSC    = ws;                 // 128000
  float* lastH  = ws + 128000;        // 128000
  float* Q      = ws + 256000;        // 256000
  float* Kt     = ws + 512000;        // 256000
  float* Ssum   = ws + 768000;        // 4000
  float* Ssq    = ws + 772000;        // 4000
  float* A1     = ws + 776000;        // 128000
  float* A2     = ws + 904000;        // 128000
  float* araw   = ws + 1032000;       // 2,000,000
  float* rcpn   = ws + 3032000;       // 4000
  float* binb   = ws + 3036000;       // 2,000,000
  float* rowdeg = ws + 5036000;       // 4000
  float* rcpdg  = ws + 5040000;       // 4000
  float* tmp    = ws + 5044000;       // 640,000
  float* finalB = ws + 5684000;       // 2,048,000   (~31 MB total)

  conv_gru_kernel<<<NODES / 4, 128, 0, stream>>>(x, conv_s_w, conv_s_b, conv_l_w,
                                                 conv_l_b, gru_Wih, gru_Whh, gru_bih,
                                                 gru_bhh, hSC, lastH);
  qk_kernel<<<(NODES * 64 + 255) / 256, 256, 0, stream>>>(hSC, WQ_w, WQ_b, WK_w, WK_b, Q, Kt);
  zero_kernel<<<(8000 + 255) / 256, 256, 0, stream>>>(Ssum, 8000);       // Ssum+Ssq contig
  zero_kernel<<<(4000 + 255) / 256, 256, 0, stream>>>(rowdeg, 4000);
  attn_stats_kernel<<<dim3(8, 32, BN), 128, 0, stream>>>(Q, Kt, Ssum, Ssq);
  assemble_kernel<<<(NODES * 192 + 255) / 256, 256, 0, stream>>>(
      hSC, Ssum, Ssq, degree, tenc_w, tenc_b, senc_w, senc_b, lastH, finalB);
  a1a2_kernel<<<(NODES * NH + 255) / 256, 256, 0, stream>>>(lastH, W1, W2, A1, A2);
  araw_kernel<<<(BN * MN * MN + 255) / 256, 256, 0, stream>>>(A1, A2, b1, Vv, bv, araw);
  colnorm_kernel<<<BN * MN, 128, 0, stream>>>(araw, rcpn);
  cgate_adj_kernel<<<dim3(8, 32, BN), 128, 0, stream>>>(araw, rcpn, Wb, adj_geo,
                                                        d_gate, degree, wb, binb, rowdeg);
  rcpdeg_kernel<<<(BN * MN + 255) / 256, 256, 0, stream>>>(rowdeg, rcpdg);
  // GNN layer 0: reads feats at col 0, writes elu output at col 160
  diffuse_kernel<<<dim3(3, 32, BN), 128, 0, stream>>>(binb, rcpdg, finalB, 0, tmp);
  dense_kernel<<<dim3(3, 32, BN), 128, 0, stream>>>(tmp, gnn_w, gnn_b, finalB, 160);
  // GNN layer 1: reads col 160, writes col 320
  diffuse_kernel<<<dim3(3, 32, BN), 128, 0, stream>>>(binb, rcpdg, finalB, 160, tmp);
  dense_kernel<<<dim3(3, 32, BN), 128, 0, stream>>>(tmp, gnn_w + DD * DD, gnn_b + DD,
                                                    finalB, 320);
  out_kernel<<<(NODES + 255) / 256, 256, 0, stream>>>(finalB, out_w, out_b, (float*)d_out);
}